// MyModel_47777216200854
// MI455X (gfx1250) — compile-verified
//
#include <hip/hip_runtime.h>
#include <cstddef>

typedef float v2f __attribute__((ext_vector_type(2)));
typedef float v8f __attribute__((ext_vector_type(8)));

#define EMBED 64
#define NN    512
#define NFEAT 9
#define CSTR  516   // curr row stride (dwords): 516 mod 64 = 4 -> conflict-free frag access
#define TSTR  68    // theta row stride (dwords): 68 mod 64 = 4

__device__ __forceinline__ v8f wmma_f32(v2f a, v2f b, v8f c) {
    // D = A(16x4 f32) x B(4x16 f32) + C(16x16 f32)
    return __builtin_amdgcn_wmma_f32_16x16x4_f32(
        /*neg_a=*/false, a, /*neg_b=*/false, b,
        /*c_mod=*/(short)0, c, /*reuse_a=*/false, /*reuse_b=*/false);
}

// K=64 matmul: acc += A(64x64, LDS, stride TSTR) * B(64 x 64-col tile of sC)
__device__ __forceinline__ void matmul64_lds(const float* __restrict__ A,
                                             const float* __restrict__ Bbuf,
                                             int col0, int koff, int ln,
                                             v8f acc[4][4]) {
    for (int kb = 0; kb < 64; kb += 4) {
        v2f a[4], b[4];
#pragma unroll
        for (int mi = 0; mi < 4; ++mi)
            a[mi] = *(const v2f*)&A[(mi * 16 + ln) * TSTR + kb + koff];
#pragma unroll
        for (int ni = 0; ni < 4; ++ni) {
            b[ni].x = Bbuf[(kb + koff) * CSTR + col0 + ni * 16];
            b[ni].y = Bbuf[(kb + koff + 1) * CSTR + col0 + ni * 16];
        }
#pragma unroll
        for (int mi = 0; mi < 4; ++mi)
#pragma unroll
            for (int ni = 0; ni < 4; ++ni)
                acc[mi][ni] = wmma_f32(a[mi], b[ni], acc[mi][ni]);
    }
}

// Write a 64x64 tile held in C-layout accumulators into sC (rows 0..63, cols col0-based)
__device__ __forceinline__ void store_tile(float* __restrict__ C, int hh, int col0,
                                           const v8f acc[4][4]) {
#pragma unroll
    for (int mi = 0; mi < 4; ++mi)
#pragma unroll
        for (int ni = 0; ni < 4; ++ni) {
            float* dst = &C[(mi * 16 + 8 * hh) * CSTR + col0 + ni * 16];
#pragma unroll
            for (int r = 0; r < 8; ++r) dst[r * CSTR] = acc[mi][ni][r];
        }
}

__global__ __launch_bounds__(256)
void graph_embed_kernel(const float* __restrict__ adj1, const float* __restrict__ fm1,
                        const float* __restrict__ adj2, const float* __restrict__ fm2,
                        const float* __restrict__ W1,  const float* __restrict__ theta,
                        const float* __restrict__ theta1, float* __restrict__ g_ws) {
    extern __shared__ float smem[];
    float* sC  = smem;                       // EMBED x CSTR   (curr, also staging)
    float* sTh = sC  + EMBED * CSTR;         // EMBED x TSTR
    float* sT1 = sTh + EMBED * TSTR;         // EMBED x TSTR
    float* sF  = sT1 + EMBED * TSTR;         // NN x NFEAT
    float* sW1 = sF  + NN * NFEAT;           // EMBED x NFEAT
    float* sG  = sW1 + EMBED * NFEAT;        // 256 reduction scratch

    const int tid   = threadIdx.x;
    const int graph = blockIdx.x >> 7;
    const int batch = blockIdx.x & 127;
    const float* adj = (graph == 0 ? adj1 : adj2) + (size_t)batch * NN * NN;
    const float* fm  = (graph == 0 ? fm1  : fm2)  + (size_t)batch * NN * NFEAT;

    for (int i = tid; i < EMBED * EMBED; i += 256) {
        int e = i >> 6, k = i & 63;
        sTh[e * TSTR + k] = theta[i];
        sT1[e * TSTR + k] = theta1[i];
    }
    for (int i = tid; i < NN * NFEAT; i += 256)    sF[i]  = fm[i];
    for (int i = tid; i < EMBED * NFEAT; i += 256) sW1[i] = W1[i];
    __syncthreads();

    const int lane = tid & 31;
    const int w    = tid >> 5;       // wave id: owns columns [64w, 64w+64)
    const int ln   = lane & 15;
    const int hh   = lane >> 4;      // lane half
    const int koff = 2 * hh;
    const int col0 = w * 64 + ln;

    const v8f vzero = {0.f, 0.f, 0.f, 0.f, 0.f, 0.f, 0.f, 0.f};

    // ---- term[e,n] = sum_f W1[e,f]*feat[n,f]  (loop-invariant, kept in C-layout regs) ----
    v8f term[4][4];
    {
        float fr[4][NFEAT];
#pragma unroll
        for (int ni = 0; ni < 4; ++ni) {
            int n = col0 + ni * 16;
#pragma unroll
            for (int f = 0; f < NFEAT; ++f) fr[ni][f] = sF[n * NFEAT + f];
        }
#pragma unroll
        for (int mi = 0; mi < 4; ++mi)
#pragma unroll
            for (int ni = 0; ni < 4; ++ni)
#pragma unroll
                for (int r = 0; r < 8; ++r) {
                    int e = mi * 16 + r + 8 * hh;
                    float s = 0.f;
#pragma unroll
                    for (int f = 0; f < NFEAT; ++f) s += sW1[e * NFEAT + f] * fr[ni][f];
                    term[mi][ni][r] = s;
                }
    }

    // ---- curr_1 = tanh(term)   (prev_0 == 0) ----
    {
        v8f c1[4][4];
#pragma unroll
        for (int mi = 0; mi < 4; ++mi)
#pragma unroll
            for (int ni = 0; ni < 4; ++ni)
#pragma unroll
                for (int r = 0; r < 8; ++r) c1[mi][ni][r] = tanhf(term[mi][ni][r]);
        store_tile(sC, hh, col0, c1);   // single full-tile store
    }
    __syncthreads();

    // per-lane adj row pointers for B fragments: B[k][i] = adj[i][k]
    const float* arow[4];
#pragma unroll
    for (int ni = 0; ni < 4; ++ni)
        arow[ni] = adj + (size_t)(col0 + ni * 16) * NN + koff;

    // ---- 4 aggregation + transform rounds (5th aggregation of reference is dead) ----
    for (int t = 0; t < 4; ++t) {
        // prev_tile[e,i] = sum_k curr[e,k] * adj[i,k]
        v8f acc[4][4];
#pragma unroll
        for (int mi = 0; mi < 4; ++mi)
#pragma unroll
            for (int ni = 0; ni < 4; ++ni) acc[mi][ni] = vzero;

        for (int kb = 0; kb < NN; kb += 4) {
            v2f a[4], b[4];
#pragma unroll
            for (int mi = 0; mi < 4; ++mi)
                a[mi] = *(const v2f*)&sC[(mi * 16 + ln) * CSTR + kb + koff];
#pragma unroll
            for (int ni = 0; ni < 4; ++ni)
                b[ni] = *(const v2f*)&arow[ni][kb];
#pragma unroll
            for (int mi = 0; mi < 4; ++mi)
#pragma unroll
                for (int ni = 0; ni < 4; ++ni)
                    acc[mi][ni] = wmma_f32(a[mi], b[ni], acc[mi][ni]);
        }
        __syncthreads();                 // all waves done reading curr
        store_tile(sC, hh, col0, acc);   // stage prev into own column slice
        __syncthreads();

        // h = relu(theta @ prev_tile)
        v8f acc2[4][4];
#pragma unroll
        for (int mi = 0; mi < 4; ++mi)
#pragma unroll
            for (int ni = 0; ni < 4; ++ni) acc2[mi][ni] = vzero;
        matmul64_lds(sTh, sC, col0, koff, ln, acc2);
#pragma unroll
        for (int mi = 0; mi < 4; ++mi)
#pragma unroll
            for (int ni = 0; ni < 4; ++ni)
#pragma unroll
                for (int r = 0; r < 8; ++r) acc2[mi][ni][r] = fmaxf(acc2[mi][ni][r], 0.f);
        store_tile(sC, hh, col0, acc2);
        __syncthreads();

        // neigh = theta1 @ h ; curr = tanh(term + neigh)
#pragma unroll
        for (int mi = 0; mi < 4; ++mi)
#pragma unroll
            for (int ni = 0; ni < 4; ++ni) acc[mi][ni] = vzero;
        matmul64_lds(sT1, sC, col0, koff, ln, acc);
#pragma unroll
        for (int mi = 0; mi < 4; ++mi)
#pragma unroll
            for (int ni = 0; ni < 4; ++ni)
#pragma unroll
                for (int r = 0; r < 8; ++r)
                    acc[mi][ni][r] = tanhf(term[mi][ni][r] + acc[mi][ni][r]);
        store_tile(sC, hh, col0, acc);
        __syncthreads();
    }

    // ---- g[e] = sum_n curr[e,n] ----
    {
        int e = tid >> 2, q = tid & 3;
        float s = 0.f;
        for (int n = q * 128; n < q * 128 + 128; ++n) s += sC[e * CSTR + n];
        sG[tid] = s;
        __syncthreads();
        if (tid < 64) {
            float g = sG[tid * 4] + sG[tid * 4 + 1] + sG[tid * 4 + 2] + sG[tid * 4 + 3];
            g_ws[(size_t)graph * EMBED * 128 + tid * 128 + batch] = g;
        }
    }
}

// out layout: [0,128)=sim, [128,8320)=g1=W2@g1raw, [8320,16512)=g2
__global__ __launch_bounds__(256)
void head_kernel(const float* __restrict__ W2, const float* __restrict__ g_ws,
                 float* __restrict__ out) {
    extern __shared__ float s2[];
    float* sW2 = s2;            // 4096
    float* sg  = sW2 + 4096;    // 2*64*128 = 16384
    float* so  = sg + 16384;    // 16384
    int tid = threadIdx.x;
    for (int i = tid; i < 4096; i += 256)  sW2[i] = W2[i];
    for (int i = tid; i < 16384; i += 256) sg[i]  = g_ws[i];
    __syncthreads();
    for (int idx = tid; idx < 16384; idx += 256) {
        int graph = idx >> 13, rem = idx & 8191;
        int e = rem >> 7, b = rem & 127;
        float s = 0.f;
#pragma unroll 8
        for (int k = 0; k < 64; ++k) s += sW2[e * 64 + k] * sg[(graph << 13) + k * 128 + b];
        so[idx] = s;
        out[128 + idx] = s;
    }
    __syncthreads();
    if (tid < 128) {
        float dot = 0.f, l1 = 0.f, l2 = 0.f;
#pragma unroll 8
        for (int e = 0; e < 64; ++e) {
            float a = so[e * 128 + tid];
            float c = so[8192 + e * 128 + tid];
            dot += a * c; l1 += a * a; l2 += c * c;
        }
        out[tid] = dot / (sqrtf(l1) * sqrtf(l2) + 1e-4f);
    }
}

extern "C" void kernel_launch(void* const* d_in, const int* in_sizes, int n_in,
                              void* d_out, int out_size, void* d_ws, size_t ws_size,
                              hipStream_t stream) {
    (void)in_sizes; (void)n_in; (void)out_size; (void)ws_size;
    const float* adj1 = (const float*)d_in[0];
    const float* fm1  = (const float*)d_in[1];
    const float* adj2 = (const float*)d_in[2];
    const float* fm2  = (const float*)d_in[3];
    const float* W1   = (const float*)d_in[4];
    const float* W2   = (const float*)d_in[5];
    const float* th   = (const float*)d_in[6];
    const float* th1  = (const float*)d_in[7];
    float* gws = (float*)d_ws;  // 2*64*128 floats

    size_t smem1 = (size_t)(EMBED * CSTR + 2 * EMBED * TSTR + NN * NFEAT +
                            EMBED * NFEAT + 256) * sizeof(float);        // ~188.7 KB
    size_t smem2 = (size_t)(4096 + 16384 + 16384) * sizeof(float);       // 144 KB

    hipFuncSetAttribute((const void*)graph_embed_kernel,
                        hipFuncAttributeMaxDynamicSharedMemorySize, (int)smem1);
    hipFuncSetAttribute((const void*)head_kernel,
                        hipFuncAttributeMaxDynamicSharedMemorySize, (int)smem2);

    graph_embed_kernel<<<256, 256, smem1, stream>>>(adj1, fm1, adj2, fm2, W1, th, th1, gws);
    head_kernel<<<1, 256, smem2, stream>>>(W2, gws, (float*)d_out);
}